// SAGE_sup_5995774346006
// MI455X (gfx1250) — compile-verified
//
#include <hip/hip_runtime.h>
#include <stdint.h>

#define NODES 50000
#define EDGES 800000
#define DIN   256
#define DH    256
#define DOUT  128
#define KDIM  256
#define LDS_STRIDE 260   // 256 + 4 pad -> conflict-free ds_load_b64 fragment reads

typedef __attribute__((ext_vector_type(2))) float v2f;
typedef __attribute__((ext_vector_type(8))) float v8f;

#if __has_builtin(__builtin_amdgcn_global_load_async_to_lds_b128) && \
    __has_builtin(__builtin_amdgcn_s_wait_asynccnt)
#define USE_ASYNC_LDS 1
typedef int v4i __attribute__((vector_size(16)));
typedef __attribute__((address_space(1))) v4i g_v4i;   // global (AS1)
typedef __attribute__((address_space(3))) v4i l_v4i;   // LDS (AS3)
#else
#define USE_ASYNC_LDS 0
#endif

__device__ __forceinline__ float fast_rcp(float x) {
#if __has_builtin(__builtin_amdgcn_rcpf)
    return __builtin_amdgcn_rcpf(x);   // v_rcp_f32
#else
    return 1.0f / x;
#endif
}

// ---------------------------------------------------------------------------
// Scatter-add aggregation: one thread per (edge, 4-float chunk), D = 256.
// 64 consecutive lanes read one contiguous 1KB node row (fully coalesced);
// scatter uses hardware f32 atomics resolving in the 192MB L2 (agg resident).
// ---------------------------------------------------------------------------
__global__ __launch_bounds__(256)
void sage_scatter_add(const float* __restrict__ x,
                      const long long* __restrict__ ei,   // [2, E] int64
                      float* __restrict__ agg,
                      float* __restrict__ cnt,            // null -> skip count
                      int E)
{
    long long t = (long long)blockIdx.x * blockDim.x + threadIdx.x;
    long long total = (long long)E * 64;
    if (t >= total) return;
    int e = (int)(t >> 6);
    int c = (int)(t & 63);
    int src = (int)ei[e];
    int dst = (int)ei[(long long)E + e];

    const float4 v = *(const float4*)(x + (long long)src * 256 + c * 4);
    float* d = agg + (long long)dst * 256 + c * 4;
    unsafeAtomicAdd(d + 0, v.x);
    unsafeAtomicAdd(d + 1, v.y);
    unsafeAtomicAdd(d + 2, v.z);
    unsafeAtomicAdd(d + 3, v.w);
    if (cnt != nullptr && c == 0) unsafeAtomicAdd(cnt + dst, 1.0f);
}

// ---------------------------------------------------------------------------
// Repack W [K, Nout] into B-fragment order: P[nt][k/4][lane][0..1], where
// lane L supplies B[k + ((L>=16)?2:0) + {0,1}][N = (nt<<4) + (L&15)].
// Makes each WMMA B-fragment a single coalesced global_load_b64.
// ---------------------------------------------------------------------------
__global__ __launch_bounds__(256)
void pack_wfrag(const float* __restrict__ W, float* __restrict__ P, int Nout)
{
    int t = blockIdx.x * blockDim.x + threadIdx.x;      // one per (nt, kq, lane)
    int total = (Nout >> 4) * (KDIM >> 2) * 32;
    if (t >= total) return;
    int lane = t & 31;
    int kq   = (t >> 5) % (KDIM >> 2);
    int nt   = (t >> 5) / (KDIM >> 2);
    int koff = (lane >> 4) << 1;
    int col  = (nt << 4) + (lane & 15);
    int kk   = (kq << 2) + koff;
    float2 v = make_float2(W[(size_t)kk * Nout + col],
                           W[(size_t)(kk + 1) * Nout + col]);
    *(float2*)(P + (size_t)t * 2) = v;
}

// ---------------------------------------------------------------------------
// Fused SAGE linear:  out = act( (agg/max(cnt,1)) @ Wl + b + xself @ Wr )
// Block = 16 output rows; A tiles staged to LDS via GLOBAL_LOAD_ASYNC_TO_LDS
// (+ s_wait_asynccnt); 8 waves each own 16-col output tiles;
// V_WMMA_F32_16X16X4_F32 over K=256, two chained WMMAs per k-step.
// NOUT is compile-time so stores use immediate offsets and the n-tile loop
// fully unrolls. Mean normalization is a per-lane v_pk_mul of the A fragment.
// ---------------------------------------------------------------------------
template <int ACT, int NOUT>   // ACT: 0 = relu, 1 = sigmoid
__global__ __launch_bounds__(256)
void sage_linear_wmma(const float* __restrict__ agg,
                      const float* __restrict__ cnt,
                      const float* __restrict__ xself,
                      const float* __restrict__ WlP,   // packed fragments
                      const float* __restrict__ WrP,   // packed fragments
                      const float* __restrict__ bias,  // [NOUT]
                      float* __restrict__ out)         // [M, NOUT]
{
    __shared__ float sMean[16][LDS_STRIDE];
    __shared__ float sSelf[16][LDS_STRIDE];

    const int tid   = threadIdx.x;
    const int mbase = blockIdx.x * 16;

    // Stage raw 16xK A tiles (normalization happens at fragment-load time).
    {
        int row = tid >> 4;            // 0..15
        int col = (tid & 15) << 4;     // 16 floats per thread
        const float* ga = agg   + (size_t)(mbase + row) * KDIM + col;
        const float* gx = xself + (size_t)(mbase + row) * KDIM + col;
#if USE_ASYNC_LDS
#pragma unroll
        for (int i = 0; i < 16; i += 4) {
            __builtin_amdgcn_global_load_async_to_lds_b128(
                (g_v4i*)(ga + i), (l_v4i*)&sMean[row][col + i], 0, 0);
            __builtin_amdgcn_global_load_async_to_lds_b128(
                (g_v4i*)(gx + i), (l_v4i*)&sSelf[row][col + i], 0, 0);
        }
        __builtin_amdgcn_s_wait_asynccnt(0);
#else
#pragma unroll
        for (int i = 0; i < 16; i += 4) {
            float4 a = *(const float4*)(ga + i);
            float4 s = *(const float4*)(gx + i);
            *(float4*)&sMean[row][col + i] = a;
            *(float4*)&sSelf[row][col + i] = s;
        }
#endif
    }
    __syncthreads();

    const int lane = tid & 31;
    const int wave = tid >> 5;
    const int r    = lane & 15;          // A row / B,C column index
    const int koff = (lane >> 4) << 1;   // 0 for lanes 0-15, 2 for lanes 16-31
    const float inv = fast_rcp(fmaxf(cnt[mbase + r], 1.0f));

    constexpr int NTILES = NOUT >> 4;
    constexpr int KQ_TOT = KDIM >> 2;
#pragma unroll
    for (int nt = wave; nt < NTILES; nt += 8) {
        const float* pbl = WlP + ((size_t)nt * KQ_TOT) * 64 + lane * 2;
        const float* pbr = WrP + ((size_t)nt * KQ_TOT) * 64 + lane * 2;
        v8f c = {};
#pragma unroll 4
        for (int k = 0; k < KDIM; k += 4) {
            v2f am = *(const v2f*)&sMean[r][k + koff];
            v2f ax = *(const v2f*)&sSelf[r][k + koff];
            v2f bl = *(const v2f*)(pbl + (k >> 2) * 64);
            v2f br = *(const v2f*)(pbr + (k >> 2) * 64);
            am *= inv;                                    // fused mean-divide
            c = __builtin_amdgcn_wmma_f32_16x16x4_f32(
                    false, am, false, bl, (short)0, c, false, false);
            c = __builtin_amdgcn_wmma_f32_16x16x4_f32(
                    false, ax, false, br, (short)0, c, false, false);
        }

        float bv = bias[(nt << 4) + r];
        const int mrow = mbase + ((lane >> 4) << 3);      // +8 for upper half
        float* p = out + (size_t)mrow * NOUT + (nt << 4) + r;
#pragma unroll
        for (int i = 0; i < 8; ++i) {
            float v = c[i] + bv;
            if (ACT == 0) v = fmaxf(v, 0.0f);
            else          v = fast_rcp(1.0f + __expf(-v));
            p[i * NOUT] = v;    // compile-time immediate offsets
        }
    }
}

// ---------------------------------------------------------------------------
extern "C" void kernel_launch(void* const* d_in, const int* in_sizes, int n_in,
                              void* d_out, int out_size, void* d_ws, size_t ws_size,
                              hipStream_t stream)
{
    const float*      x   = (const float*)d_in[0];
    const long long*  ei  = (const long long*)d_in[1];   // int64 [2, E]
    const float*      W1l = (const float*)d_in[2];
    const float*      b1  = (const float*)d_in[3];
    const float*      W1r = (const float*)d_in[4];
    const float*      W2l = (const float*)d_in[5];
    const float*      b2  = (const float*)d_in[6];
    const float*      W2r = (const float*)d_in[7];
    float*            out = (float*)d_out;

    // Workspace: cnt | agg [N,256] | h [N,256] | packed weights (~103.5 MB)
    char*  ws      = (char*)d_ws;
    size_t off_agg = ((size_t)NODES * 4 + 1023) & ~(size_t)1023;
    size_t off_h   = off_agg + (size_t)NODES * 256 * 4;
    size_t off_p1l = off_h   + (size_t)NODES * 256 * 4;
    size_t off_p1r = off_p1l + (size_t)KDIM * DH * 4;
    size_t off_p2l = off_p1r + (size_t)KDIM * DH * 4;
    size_t off_p2r = off_p2l + (size_t)KDIM * DOUT * 4;
    float* cnt  = (float*)ws;
    float* agg  = (float*)(ws + off_agg);
    float* h    = (float*)(ws + off_h);
    float* W1lP = (float*)(ws + off_p1l);
    float* W1rP = (float*)(ws + off_p1r);
    float* W2lP = (float*)(ws + off_p2l);
    float* W2rP = (float*)(ws + off_p2r);

    const long long scat_threads = (long long)EDGES * 64;
    const int scat_blocks = (int)((scat_threads + 255) / 256);

    // Weight fragment repack (tiny: <1.6 MB total)
    {
        int tot256 = (DH   >> 4) * (KDIM >> 2) * 32;   // 32768
        int tot128 = (DOUT >> 4) * (KDIM >> 2) * 32;   // 16384
        pack_wfrag<<<(tot256 + 255) / 256, 256, 0, stream>>>(W1l, W1lP, DH);
        pack_wfrag<<<(tot256 + 255) / 256, 256, 0, stream>>>(W1r, W1rP, DH);
        pack_wfrag<<<(tot128 + 255) / 256, 256, 0, stream>>>(W2l, W2lP, DOUT);
        pack_wfrag<<<(tot128 + 255) / 256, 256, 0, stream>>>(W2r, W2rP, DOUT);
    }

    // ---- layer 1 ----
    (void)hipMemsetAsync(cnt, 0, (size_t)NODES * 4, stream);
    (void)hipMemsetAsync(agg, 0, (size_t)NODES * DIN * 4, stream);
    sage_scatter_add<<<scat_blocks, 256, 0, stream>>>(x, ei, agg, cnt, EDGES);
    sage_linear_wmma<0, DH><<<NODES / 16, 256, 0, stream>>>(
        agg, cnt, x, W1lP, W1rP, b1, h);

    // ---- layer 2 ----
    (void)hipMemsetAsync(agg, 0, (size_t)NODES * DH * 4, stream);
    sage_scatter_add<<<scat_blocks, 256, 0, stream>>>(h, ei, agg, nullptr, EDGES);
    sage_linear_wmma<1, DOUT><<<NODES / 16, 256, 0, stream>>>(
        agg, cnt, h, W2lP, W2rP, b2, out);
}